// Noise_47407849013270
// MI455X (gfx1250) — compile-verified
//
#include <hip/hip_runtime.h>
#include <hip/hip_bf16.h>
#include <stdint.h>

// ---- vector / descriptor types -------------------------------------------
typedef __attribute__((ext_vector_type(4))) float        fv4;
typedef __attribute__((ext_vector_type(4))) unsigned int u32x4;
typedef __attribute__((ext_vector_type(4))) int          i32x4;
typedef __attribute__((ext_vector_type(8))) int          i32x8;

#define TILE_ELEMS 4096            // floats per tile (16 KB)
#define TILE_VEC4  (TILE_ELEMS/4)  // 1024 float4 per tile
#define BLOCK      256             // 8 wave32 waves

// ---- TDM: issue an async tile load (global -> LDS) -----------------------
// D# group0/group1 packed per CDNA5 ISA 8.3/8.4 (2D degenerate: one row of
// TILE_ELEMS fp32 elements). tensor_dim0 = remaining elements => OOB reads
// past the end of the array return zero instead of faulting.
__device__ __forceinline__ void tdm_load_tile(const float* src,
                                              long long tileStartElem,
                                              long long totalElems,
                                              unsigned ldsByteOff) {
    unsigned long long ga = (unsigned long long)(src + tileStartElem);
    long long remaining   = totalElems - tileStartElem;
    unsigned td0 = (remaining > 0xFFFFFFFFll) ? 0xFFFFFFFFu : (unsigned)remaining;

    u32x4 g0;
    g0.x = 1u;                                   // count=1, user mode, no gather
    g0.y = ldsByteOff;                           // lds_addr [63:32]
    g0.z = (unsigned)(ga & 0xFFFFFFFFull);       // global_addr [95:64]
    g0.w = (unsigned)((ga >> 32) & 0x01FFFFFFull) | (2u << 30); // addr hi + type=2

    i32x8 g1;
    g1.s0 = (int)(2u << 16);                     // workgroup_mask=0, data_size=2 (4B)
    g1.s1 = (int)((td0 & 0xFFFFu) << 16);        // tensor_dim0 lo16 in [63:48]
    g1.s2 = (int)((td0 >> 16) | (1u << 16));     // tensor_dim0 hi16, tensor_dim1=1
    g1.s3 = (int)((unsigned)TILE_ELEMS << 16);   // tensor_dim1 hi=0, tile_dim0
    g1.s4 = 1;                                   // tile_dim1=1, tile_dim2=0
    g1.s5 = TILE_ELEMS;                          // tensor_dim0_stride lo32
    g1.s6 = 0;                                   // stride0 hi16, stride1 lo16
    g1.s7 = 0;                                   // stride1 hi32

    i32x4 gz = {0, 0, 0, 0};                     // groups 2/3 unused (<=2D)
#if defined(__clang_major__) && (__clang_major__ >= 23)
    i32x8 gz8 = {0, 0, 0, 0, 0, 0, 0, 0};
    __builtin_amdgcn_tensor_load_to_lds(g0, g1, gz, gz, gz8, 0);
#else
    __builtin_amdgcn_tensor_load_to_lds(g0, g1, gz, gz, 0);
#endif
}

// ---- Philox4x32-10 -------------------------------------------------------
__device__ __forceinline__ void philox4x32(unsigned c0, unsigned c1,
                                           unsigned c2, unsigned c3,
                                           unsigned k0, unsigned k1,
                                           unsigned& o0, unsigned& o1,
                                           unsigned& o2, unsigned& o3) {
#pragma unroll
    for (int r = 0; r < 10; ++r) {
        unsigned hi0 = __umulhi(0xD2511F53u, c0);
        unsigned lo0 = 0xD2511F53u * c0;
        unsigned hi1 = __umulhi(0xCD9E8D57u, c2);
        unsigned lo1 = 0xCD9E8D57u * c2;
        unsigned n0 = hi1 ^ c1 ^ k0;
        unsigned n1 = lo1;
        unsigned n2 = hi0 ^ c3 ^ k1;
        unsigned n3 = lo0;
        c0 = n0; c1 = n1; c2 = n2; c3 = n3;
        k0 += 0x9E3779B9u; k1 += 0xBB67AE85u;
    }
    o0 = c0; o1 = c1; o2 = c2; o3 = c3;
}

// Box-Muller: two u32 -> two N(0,1)
__device__ __forceinline__ void box_muller(unsigned a, unsigned b,
                                           float& n0, float& n1) {
    // u1 in (0,1], u2 in [0,1)
    float u1 = fmaf((float)(a >> 8), 0x1p-24f, 0x1p-25f);
    float u2 = (float)(b >> 8) * 0x1p-24f;
    float r  = __fsqrt_rn(-2.0f * __logf(u1));
    float s, c;
    __sincosf(6.28318530717958647692f * u2, &s, &c);
    n0 = r * c;
    n1 = r * s;
}

// ---- kernel --------------------------------------------------------------
__global__ void __launch_bounds__(BLOCK)
noise_add_kernel(const float* __restrict__ in, float* __restrict__ out,
                 long long N, int tiles) {
    __shared__ fv4 lds[2][TILE_VEC4];            // 2 x 16 KB double buffer

    const unsigned ldsOff0 = (unsigned)(uintptr_t)&lds[0][0];
    const unsigned ldsOff1 = (unsigned)(uintptr_t)&lds[1][0];

    int t   = (int)blockIdx.x;
    int cur = 0;

    // prologue: wave 0 kicks off DMA for the first tile
    if (threadIdx.x < 32) {
        if (t < tiles)
            tdm_load_tile(in, (long long)t * TILE_ELEMS, N, ldsOff0);
    }

    for (; t < tiles; t += (int)gridDim.x) {
        int tn = t + (int)gridDim.x;
        if (threadIdx.x < 32) {
            if (tn < tiles) {
                // issue next tile into the other buffer, then wait until only
                // that one is outstanding => current tile's DMA is complete.
                tdm_load_tile(in, (long long)tn * TILE_ELEMS, N,
                              cur ? ldsOff0 : ldsOff1);
                __builtin_amdgcn_s_wait_tensorcnt(1);
            } else {
                __builtin_amdgcn_s_wait_tensorcnt(0);
            }
        }
        __syncthreads();                         // tile data visible to all waves

        long long tileVecBase = ((long long)t * TILE_ELEMS) >> 2;
#pragma unroll
        for (int k = 0; k < 4; ++k) {
            int v = (int)threadIdx.x + k * BLOCK;        // 0..1023
            long long gvec = tileVecBase + v;            // global float4 index
            if ((gvec << 2) < N) {
                fv4 x = lds[cur][v];

                unsigned r0, r1, r2, r3;
                philox4x32((unsigned)gvec,
                           (unsigned)((unsigned long long)gvec >> 32),
                           0x2E7035C5u, 0x1BD11BDAu,   // stream constants
                           42u, 0u,                    // "key(42)"
                           r0, r1, r2, r3);

                float n0, n1, n2, n3;
                box_muller(r0, r1, n0, n1);
                box_muller(r2, r3, n2, n3);

                fv4 o;
                o.x = fmaf(0.1f, n0, x.x);
                o.y = fmaf(0.1f, n1, x.y);
                o.z = fmaf(0.1f, n2, x.z);
                o.w = fmaf(0.1f, n3, x.w);
                // streamed once, never re-read: non-temporal 128-bit store
                __builtin_nontemporal_store(o, ((fv4*)out) + gvec);
            }
        }
        __syncthreads();                         // readers done before overwrite
        cur ^= 1;
    }
}

// ---- host-side launch ----------------------------------------------------
extern "C" void kernel_launch(void* const* d_in, const int* in_sizes, int n_in,
                              void* d_out, int out_size, void* d_ws, size_t ws_size,
                              hipStream_t stream) {
    const float* in = (const float*)d_in[0];
    float* out      = (float*)d_out;
    long long N     = (long long)in_sizes[0];

    int tiles = (int)((N + TILE_ELEMS - 1) / TILE_ELEMS);   // 12288 for this shape
    int grid  = tiles < 3072 ? tiles : 3072;                // ~4 tiles/block pipeline

    noise_add_kernel<<<grid, BLOCK, 0, stream>>>(in, out, N, tiles);
}